// EmbeddingProposal_49538152792249
// MI455X (gfx1250) — compile-verified
//
#include <hip/hip_runtime.h>
#include <math.h>

typedef __attribute__((ext_vector_type(2))) float v2f;
typedef __attribute__((ext_vector_type(8))) float v8f;

#define KP 128   // particles
#define TT 64    // subtrees
#define DD 128   // embedding dim
#define HH 512   // hidden dim

// workspace layout (float offsets)
#define WS_W      0                        // K*T*T merge log-weights (524288)
#define WS_NORM   (WS_W + KP*TT*TT)        // K*T squared norms (8192)
#define WS_X      (WS_NORM + KP*TT)        // K*256 concat children (32768)
#define WS_H      (WS_X + KP*256)          // K*512 hidden (65536)
#define WS_LMP    (WS_H + KP*HH)           // K log merge prob
#define WS_IDX1   (WS_LMP + KP)            // K ints
#define WS_IDX2   (WS_IDX1 + KP)           // K ints

// output layout (floats): idx1,idx2,branch1,branch2,emb,log_v_plus,log_v_minus
#define OUT_IDX1 0
#define OUT_IDX2 (KP)
#define OUT_B1   (2*KP)
#define OUT_B2   (3*KP)
#define OUT_EMB  (4*KP)
#define OUT_LVP  (4*KP + KP*DD)
#define OUT_LVM  (5*KP + KP*DD)

// ---------------------------------------------------------------- norms
__global__ void ep_norms(const float* __restrict__ E, float* __restrict__ norms) {
    int idx = blockIdx.x * blockDim.x + threadIdx.x;   // K*T = 8192 threads
    const float* p = E + idx * DD;
    float s = 0.f;
    #pragma unroll 8
    for (int d = 0; d < DD; ++d) { float v = p[d]; s += v * v; }
    norms[idx] = s;
}

// ------------------------------------------- Gram via WMMA -> log-weights
// one wave per 16x16 tile of the 64x64 Gram matrix; K*16 tiles total
__global__ void ep_gram(const float* __restrict__ E, const float* __restrict__ norms,
                        float* __restrict__ w) {
    int warp = threadIdx.x >> 5;
    unsigned l = threadIdx.x & 31;
    int tile = blockIdx.x * 8 + warp;          // 0 .. K*16-1
    int k   = tile >> 4;
    int t16 = tile & 15;
    int i0 = (t16 >> 2) * 16;                  // row tile
    int j0 = (t16 & 3) * 16;                   // col tile
    int lm = (int)(l & 15);
    int lh = (int)(l >> 4);                    // 0 or 1
    const float* Ek = E + (size_t)k * TT * DD;
    const float* arow = Ek + (i0 + lm) * DD;   // A: rows i0..i0+15
    const float* brow = Ek + (j0 + lm) * DD;   // B^T: rows j0..j0+15

    v8f c = {0.f,0.f,0.f,0.f,0.f,0.f,0.f,0.f};
    for (int kb = 0; kb < DD; kb += 4) {
        v2f a, b;
        a.x = arow[kb + lh*2];       // A[m= l%16][k = kb + lh*2 + {0,1}]
        a.y = arow[kb + lh*2 + 1];
        b.x = brow[kb + lh];         // B[k = kb+lh][n = l%16]   (B = E^T)
        b.y = brow[kb + 2 + lh];     // B[k = kb+2+lh][n]
        c = __builtin_amdgcn_wmma_f32_16x16x4_f32(false, a, false, b,
                                                  (short)0, c, false, false);
    }

    int colg = j0 + lm;
    float nj = norms[k*TT + colg];
    float* wk = w + (size_t)k * TT * TT;
    #pragma unroll
    for (int v = 0; v < 8; ++v) {
        int rowg = i0 + v + lh*8;              // C/D layout: M = v + (lane>=16)*8
        float ni = norms[k*TT + rowg];
        float d2 = ni + nj - 2.0f * c[v];
        float val = -sqrtf(fmaxf(d2, 0.0f) + 1e-12f);
        if (rowg == colg) val = -__builtin_inff();   // forbid self-merge
        wk[rowg * TT + colg] = val;
    }
}

// --------------------- per-particle: gumbel argmax + logsumexp + gather X
__global__ void ep_sample(const float* __restrict__ w, const float* __restrict__ g,
                          const int* __restrict__ lc, const int* __restrict__ Np,
                          const float* __restrict__ E, float* __restrict__ X,
                          float* __restrict__ lmp, int* __restrict__ i1w,
                          int* __restrict__ i2w, float* __restrict__ out) {
    int k = blockIdx.x;
    int t = threadIdx.x;                       // 256 threads
    __shared__ float s_v[256];
    __shared__ int   s_i[256];
    __shared__ float s_m[256];
    __shared__ int   sb_i1, sb_i2, sb_sample;
    __shared__ float sb_mw;

    const float* wk = w + (size_t)k * TT * TT;
    const float* gk = g + (size_t)k * TT * TT;

    float best = -__builtin_inff(); int bi = 0;
    float mw   = -__builtin_inff();
    for (int j = 0; j < 16; ++j) {
        int e = t + j * 256;
        float wv = wk[e];
        float tv = wv + gk[e];
        if (tv > best) { best = tv; bi = e; }
        if (wv > mw) mw = wv;
    }
    s_v[t] = best; s_i[t] = bi; s_m[t] = mw;
    __syncthreads();
    for (int s = 128; s > 0; s >>= 1) {
        if (t < s) {
            if (s_v[t+s] > s_v[t]) { s_v[t] = s_v[t+s]; s_i[t] = s_i[t+s]; }
            if (s_m[t+s] > s_m[t]) s_m[t] = s_m[t+s];
        }
        __syncthreads();
    }
    if (t == 0) { sb_sample = s_i[0]; sb_mw = s_m[0]; }
    __syncthreads();

    float maxw = sb_mw;
    float sum = 0.f;
    for (int j = 0; j < 16; ++j) {
        int e = t + j * 256;
        sum += expf(wk[e] - maxw);             // exp(-inf - m) = 0 on diagonal
    }
    s_v[t] = sum;
    __syncthreads();
    for (int s = 128; s > 0; s >>= 1) { if (t < s) s_v[t] += s_v[t+s]; __syncthreads(); }

    if (t == 0) {
        int sample = sb_sample;
        int i1 = sample >> 6, i2 = sample & 63;
        sb_i1 = i1; sb_i2 = i2;
        float lse = maxw + logf(s_v[0]);
        lmp[k] = wk[sample] + 0.69314718055994531f - lse;   // + log(2)
        i1w[k] = i1; i2w[k] = i2;
        out[OUT_IDX1 + k] = (float)i1;
        out[OUT_IDX2 + k] = (float)i2;
        int ones = 0;
        for (int tt = 0; tt < TT; ++tt) ones += (lc[k*TT + tt] == 1);
        ones -= (lc[k*TT + i1] == 1);
        ones -= (lc[k*TT + i2] == 1);
        out[OUT_LVM + k] = logf((float)(Np[0] - ones));
    }
    __syncthreads();
    int i1 = sb_i1, i2 = sb_i2;
    if (t < 128) X[k*256 + t]           = E[(size_t)k*TT*DD + i1*DD + t];
    else         X[k*256 + 128 + (t-128)] = E[(size_t)k*TT*DD + i2*DD + (t-128)];
}

// ------------------------------------- GEMM1: X(128x256)@W1(256x512)+b1, relu
__global__ void ep_mlp1(const float* __restrict__ X, const float* __restrict__ W1,
                        const float* __restrict__ b1, float* __restrict__ Hm) {
    int warp = threadIdx.x >> 5;
    unsigned l = threadIdx.x & 31;
    int tile = blockIdx.x * 8 + warp;          // 0..255 (8 M-tiles x 32 N-tiles)
    int i0 = (tile >> 5) * 16;
    int j0 = (tile & 31) * 16;
    int lm = (int)(l & 15), lh = (int)(l >> 4);
    const float* arow = X + (i0 + lm) * 256;

    v8f c = {0.f,0.f,0.f,0.f,0.f,0.f,0.f,0.f};
    for (int kb = 0; kb < 256; kb += 4) {
        v2f a, b;
        a.x = arow[kb + lh*2];
        a.y = arow[kb + lh*2 + 1];
        b.x = W1[(kb + lh)   * HH + j0 + lm];
        b.y = W1[(kb + 2+lh) * HH + j0 + lm];
        c = __builtin_amdgcn_wmma_f32_16x16x4_f32(false, a, false, b,
                                                  (short)0, c, false, false);
    }
    float bias = b1[j0 + lm];
    #pragma unroll
    for (int v = 0; v < 8; ++v) {
        int row = i0 + v + lh*8;
        Hm[row * HH + j0 + lm] = fmaxf(c[v] + bias, 0.0f);
    }
}

// ------------------------------------- GEMM2: H(128x512)@W2(512x128)+b2 -> emb
__global__ void ep_mlp2(const float* __restrict__ Hm, const float* __restrict__ W2,
                        const float* __restrict__ b2, float* __restrict__ out) {
    int warp = threadIdx.x >> 5;
    unsigned l = threadIdx.x & 31;
    int tile = blockIdx.x * 8 + warp;          // 0..63 (8 M-tiles x 8 N-tiles)
    int i0 = (tile >> 3) * 16;
    int j0 = (tile & 7)  * 16;
    int lm = (int)(l & 15), lh = (int)(l >> 4);
    const float* arow = Hm + (i0 + lm) * HH;

    v8f c = {0.f,0.f,0.f,0.f,0.f,0.f,0.f,0.f};
    for (int kb = 0; kb < HH; kb += 4) {
        v2f a, b;
        a.x = arow[kb + lh*2];
        a.y = arow[kb + lh*2 + 1];
        b.x = W2[(kb + lh)   * DD + j0 + lm];
        b.y = W2[(kb + 2+lh) * DD + j0 + lm];
        c = __builtin_amdgcn_wmma_f32_16x16x4_f32(false, a, false, b,
                                                  (short)0, c, false, false);
    }
    float bias = b2[j0 + lm];
    #pragma unroll
    for (int v = 0; v < 8; ++v) {
        int row = i0 + v + lh*8;
        out[OUT_EMB + row * DD + j0 + lm] = c[v] + bias;
    }
}

// -------------------------- branch lengths, priors, log_v_plus (per particle)
__global__ void ep_branch(const float* __restrict__ E, const int* __restrict__ i1w,
                          const int* __restrict__ i2w, const float* __restrict__ lmp,
                          const float* __restrict__ u1, const float* __restrict__ u2,
                          float* __restrict__ out) {
    int k = blockIdx.x, d = threadIdx.x;       // 128 threads
    __shared__ float s1[128], s2[128];
    float emb = out[OUT_EMB + k*DD + d];
    float c1  = E[(size_t)k*TT*DD + i1w[k]*DD + d];
    float c2  = E[(size_t)k*TT*DD + i2w[k]*DD + d];
    float f1 = c1 - emb, f2 = c2 - emb;
    s1[d] = f1*f1; s2[d] = f2*f2;
    __syncthreads();
    for (int s = 64; s > 0; s >>= 1) {
        if (d < s) { s1[d] += s1[d+s]; s2[d] += s2[d+s]; }
        __syncthreads();
    }
    if (d == 0) {
        float d1 = sqrtf(s1[0] + 1e-12f);
        float d2 = sqrtf(s2[0] + 1e-12f);
        float r1 = 1.0f / (d1 + 1e-4f);
        float r2 = 1.0f / (d2 + 1e-4f);
        float b1v = -(1.0f / r1) * logf(u1[k]);
        float b2v = -(1.0f / r2) * logf(u2[k]);
        float lp1 = logf(r1) - r1 * b1v;
        float lp2 = logf(r2) - r2 * b2v;
        out[OUT_B1 + k]  = b1v;
        out[OUT_B2 + k]  = b2v;
        out[OUT_LVP + k] = lmp[k] + lp1 + lp2;
    }
}

extern "C" void kernel_launch(void* const* d_in, const int* in_sizes, int n_in,
                              void* d_out, int out_size, void* d_ws, size_t ws_size,
                              hipStream_t stream) {
    const int*   Np = (const int*)d_in[0];
    const int*   lc = (const int*)d_in[1];
    const float* E  = (const float*)d_in[2];
    // d_in[3] (log_felsensteins) and d_in[4] (site_positions) are unused by the reference
    const float* W1 = (const float*)d_in[5];
    const float* b1 = (const float*)d_in[6];
    const float* W2 = (const float*)d_in[7];
    const float* b2 = (const float*)d_in[8];
    const float* g  = (const float*)d_in[9];
    const float* u1 = (const float*)d_in[10];
    const float* u2 = (const float*)d_in[11];
    float* out = (float*)d_out;
    float* ws  = (float*)d_ws;

    float* w     = ws + WS_W;
    float* norms = ws + WS_NORM;
    float* X     = ws + WS_X;
    float* Hm    = ws + WS_H;
    float* lmp   = ws + WS_LMP;
    int*   i1w   = (int*)(ws + WS_IDX1);
    int*   i2w   = (int*)(ws + WS_IDX2);

    ep_norms <<<(KP*TT)/256, 256, 0, stream>>>(E, norms);
    ep_gram  <<<(KP*16)/8,   256, 0, stream>>>(E, norms, w);
    ep_sample<<<KP,          256, 0, stream>>>(w, g, lc, Np, E, X, lmp, i1w, i2w, out);
    ep_mlp1  <<<(8*32)/8,    256, 0, stream>>>(X, W1, b1, Hm);
    ep_mlp2  <<<(8*8)/8,     256, 0, stream>>>(Hm, W2, b2, out);
    ep_branch<<<KP,          DD,  0, stream>>>(E, i1w, i2w, lmp, u1, u2, out);
}